// NeuralNetwork_48490180772349
// MI455X (gfx1250) — compile-verified
//
#include <hip/hip_runtime.h>
#include <cstdint>

// ---------------- problem constants (from the reference) ----------------
#define N_POINTS   200
#define BATCH      32768
#define LPR        4                      // lanes cooperating on one ray
#define BLOCK      512                    // 16 wave32s per workgroup
#define RAYS_PER_BLOCK (BLOCK / LPR)      // 128
#define NBLOCKS    (BATCH / RAYS_PER_BLOCK) // 256

#define R0 8
#define R1 16
#define R2 32
#define S0 729     /* (R0+1)^3 */
#define S1 4913    /* (R1+1)^3 */
#define S2 35937   /* (R2+1)^3 */
#define LDS_FLOATS (S0 + S1 + S2)         // 41579 floats
#define LDS_BYTES  (LDS_FLOATS * 4)       // 166,316 B  (needs CDNA5 320KB LDS)

#define AS1 __attribute__((address_space(1)))
#define AS3 __attribute__((address_space(3)))

#ifndef __has_builtin
#define __has_builtin(x) 0
#endif
#if __has_builtin(__builtin_amdgcn_global_load_async_to_lds_b32)
#define HAVE_ASYNC_LDS 1
#else
#define HAVE_ASYNC_LDS 0
#endif
#if __has_builtin(__builtin_amdgcn_s_wait_asynccnt)
#define HAVE_WAIT_ASYNC 1
#else
#define HAVE_WAIT_ASYNC 0
#endif

__device__ __forceinline__ float lerpf_(float a, float b, float t) { return fmaf(t, b - a, a); }
__device__ __forceinline__ float sigmoidf_(float x) { return 1.0f / (1.0f + __expf(-x)); }
__device__ __forceinline__ float clamp01_(float v) { return fminf(fmaxf(v, 0.0f), 1.0f); }

// Trilinear interpolation of the compacted channel-0 plane held in LDS.
// u01* are the already-clamped unit-cube coordinates (shared across grids).
__device__ __forceinline__ float interp_ch0(const float* __restrict__ base,
                                            float u01x, float u01y, float u01z,
                                            int R) {
  const float Rf = (float)R;
  float ux = u01x * Rf, uy = u01y * Rf, uz = u01z * Rf;
  int ix = (int)ux; ix = (ix > R - 1) ? (R - 1) : ix;
  int iy = (int)uy; iy = (iy > R - 1) ? (R - 1) : iy;
  int iz = (int)uz; iz = (iz > R - 1) ? (R - 1) : iz;
  float fx = ux - (float)ix, fy = uy - (float)iy, fz = uz - (float)iz;
  const int Rp1 = R + 1;
  const float* p = base + (ix * Rp1 + iy) * Rp1 + iz;   // LDS gathers
  const float* q = p + Rp1 * Rp1;
  float v000 = p[0],   v001 = p[1];
  float v010 = p[Rp1], v011 = p[Rp1 + 1];
  float v100 = q[0],   v101 = q[1];
  float v110 = q[Rp1], v111 = q[Rp1 + 1];
  float a = lerpf_(v000, v001, fz);
  float b = lerpf_(v010, v011, fz);
  float c = lerpf_(v100, v101, fz);
  float d = lerpf_(v110, v111, fz);
  return lerpf_(lerpf_(a, b, fy), lerpf_(c, d, fy), fx);
}

// Trilinear interpolation of channels 1..3 straight from the interleaved
// global table (float4 rows, L2-resident). Used once per ray.
__device__ __forceinline__ void interp_ch123(const float* __restrict__ tab,
                                             float u01x, float u01y, float u01z,
                                             int R, float* __restrict__ o3) {
  const float Rf = (float)R;
  float ux = u01x * Rf, uy = u01y * Rf, uz = u01z * Rf;
  int ix = (int)ux; ix = (ix > R - 1) ? (R - 1) : ix;
  int iy = (int)uy; iy = (iy > R - 1) ? (R - 1) : iy;
  int iz = (int)uz; iz = (iz > R - 1) ? (R - 1) : iz;
  float fx = ux - (float)ix, fy = uy - (float)iy, fz = uz - (float)iz;
  const int Rp1 = R + 1;
  const float4* t = (const float4*)tab;
  int l0 = (ix * Rp1 + iy) * Rp1 + iz;
  int l1 = l0 + Rp1 * Rp1;
  float4 v000 = t[l0],       v001 = t[l0 + 1];
  float4 v010 = t[l0 + Rp1], v011 = t[l0 + Rp1 + 1];
  float4 v100 = t[l1],       v101 = t[l1 + 1];
  float4 v110 = t[l1 + Rp1], v111 = t[l1 + Rp1 + 1];
  {
    float a = lerpf_(v000.y, v001.y, fz), b = lerpf_(v010.y, v011.y, fz);
    float c = lerpf_(v100.y, v101.y, fz), d = lerpf_(v110.y, v111.y, fz);
    o3[0] = lerpf_(lerpf_(a, b, fy), lerpf_(c, d, fy), fx);
  }
  {
    float a = lerpf_(v000.z, v001.z, fz), b = lerpf_(v010.z, v011.z, fz);
    float c = lerpf_(v100.z, v101.z, fz), d = lerpf_(v110.z, v111.z, fz);
    o3[1] = lerpf_(lerpf_(a, b, fy), lerpf_(c, d, fy), fx);
  }
  {
    float a = lerpf_(v000.w, v001.w, fz), b = lerpf_(v010.w, v011.w, fz);
    float c = lerpf_(v100.w, v101.w, fz), d = lerpf_(v110.w, v111.w, fz);
    o3[2] = lerpf_(lerpf_(a, b, fy), lerpf_(c, d, fy), fx);
  }
}

__launch_bounds__(BLOCK)
__global__ void nerf_ray_march_kernel(const float* __restrict__ x,
                                      const float* __restrict__ g0,
                                      const float* __restrict__ g1,
                                      const float* __restrict__ g2,
                                      const float* __restrict__ w_label,
                                      const float* __restrict__ b_label,
                                      const float* __restrict__ w_rgb,
                                      const float* __restrict__ b_rgb,
                                      float* __restrict__ out) {
  extern __shared__ float sgrid[];   // [S0 | S1 | S2] compacted channel-0 planes

  // ---- stage channel-0 of all three grids into LDS (stride-4 de-interleave)
  // via the CDNA5 async global->LDS path (tracked by ASYNCcnt). ----
  {
    const int tid = threadIdx.x;
#if HAVE_ASYNC_LDS
    for (int i = tid; i < S0; i += BLOCK)
      __builtin_amdgcn_global_load_async_to_lds_b32(
          (AS1 int*)(g0 + 4 * i), (AS3 int*)(sgrid + i), 0, 0);
    for (int i = tid; i < S1; i += BLOCK)
      __builtin_amdgcn_global_load_async_to_lds_b32(
          (AS1 int*)(g1 + 4 * i), (AS3 int*)(sgrid + S0 + i), 0, 0);
    for (int i = tid; i < S2; i += BLOCK)
      __builtin_amdgcn_global_load_async_to_lds_b32(
          (AS1 int*)(g2 + 4 * i), (AS3 int*)(sgrid + S0 + S1 + i), 0, 0);
#if HAVE_WAIT_ASYNC
    __builtin_amdgcn_s_wait_asynccnt(0);
#else
    asm volatile("s_wait_asynccnt 0" ::: "memory");
#endif
#else
    for (int i = tid; i < S0; i += BLOCK) sgrid[i]            = g0[4 * i];
    for (int i = tid; i < S1; i += BLOCK) sgrid[S0 + i]       = g1[4 * i];
    for (int i = tid; i < S2; i += BLOCK) sgrid[S0 + S1 + i]  = g2[4 * i];
#endif
    __syncthreads();
  }

  const int ray = blockIdx.x * RAYS_PER_BLOCK + (threadIdx.x >> 2);
  const int sub = threadIdx.x & (LPR - 1);
  if (ray >= BATCH) return;

  // ---- per-ray geometry (each of the 4 lanes redundantly; negligible) ----
  float4 xv = ((const float4*)x)[ray];
  float s0, c0, s1, c1, s2, c2, s3, c3;
  __sincosf(xv.x, &s0, &c0);
  __sincosf(xv.y, &s1, &c1);
  __sincosf(xv.z, &s2, &c2);
  __sincosf(xv.w, &s3, &c3);
  const float p1x = s0 * c1, p1y = s0 * s1, p1z = c0;
  const float p2x = s2 * c3, p2y = s2 * s3, p2z = c2;
  // Fold the (p+1)*0.5 normalization into the ray: u01 = clamp(q1 + qd*t).
  const float q1x = (p1x + 1.0f) * 0.5f, qdx = (p2x - p1x) * 0.5f;
  const float q1y = (p1y + 1.0f) * 0.5f, qdy = (p2y - p1y) * 0.5f;
  const float q1z = (p1z + 1.0f) * 0.5f, qdz = (p2z - p1z) * 0.5f;

  const float wl0 = w_label[0], wl1 = w_label[1], wl2 = w_label[2], bl = b_label[0];

  const float* L0 = sgrid;
  const float* L1 = sgrid + S0;
  const float* L2 = sgrid + S0 + S1;

  // ---- hot loop: 50 points per lane, labels only (channel 0, LDS gathers) ----
  float maxLogit = -3.4e38f;
  int firstHit = N_POINTS;  // sentinel: "no hit"
  #pragma unroll 2
  for (int i = sub; i < N_POINTS; i += LPR) {
    const float t = (float)i * (1.0f / 199.0f);
    const float ux = clamp01_(fmaf(qdx, t, q1x));
    const float uy = clamp01_(fmaf(qdy, t, q1y));
    const float uz = clamp01_(fmaf(qdz, t, q1z));
    const float l0v = interp_ch0(L0, ux, uy, uz, R0);
    const float l1v = interp_ch0(L1, ux, uy, uz, R1);
    const float l2v = interp_ch0(L2, ux, uy, uz, R2);
    const float logit = fmaf(l0v, wl0, fmaf(l1v, wl1, fmaf(l2v, wl2, bl)));
    maxLogit = fmaxf(maxLogit, logit);
    // branchless first-hit (i is increasing within this lane)
    const int cand = (logit > 0.0f) ? i : N_POINTS;
    firstHit = (cand < firstHit) ? cand : firstHit;
  }

  // ---- reduce across the 4 lanes of this ray (xor 1,2 stays in the group) ----
  #pragma unroll
  for (int m = 1; m < LPR; m <<= 1) {
    maxLogit = fmaxf(maxLogit, __shfl_xor(maxLogit, m));
    int oth  = __shfl_xor(firstHit, m);
    firstHit = (oth < firstHit) ? oth : firstHit;
  }

  // ---- lane 0: outputs (sigmoid is monotonic -> hits = sigmoid(max logit)) ----
  if (sub == 0) {
    out[ray] = sigmoidf_(maxLogit);

    const int idx = (firstHit >= N_POINTS) ? 0 : firstHit;  // argmax of all-zero = 0
    const float t = (float)idx * (1.0f / 199.0f);
    const float ux = clamp01_(fmaf(qdx, t, q1x));
    const float uy = clamp01_(fmaf(qdy, t, q1y));
    const float uz = clamp01_(fmaf(qdz, t, q1z));

    float f[9];
    interp_ch123(g0, ux, uy, uz, R0, f + 0);
    interp_ch123(g1, ux, uy, uz, R1, f + 3);
    interp_ch123(g2, ux, uy, uz, R2, f + 6);

    float rr = b_rgb[0], gg = b_rgb[1], bb = b_rgb[2];
    #pragma unroll
    for (int k = 0; k < 9; ++k) {
      rr = fmaf(f[k], w_rgb[3 * k + 0], rr);
      gg = fmaf(f[k], w_rgb[3 * k + 1], gg);
      bb = fmaf(f[k], w_rgb[3 * k + 2], bb);
    }
    float* orgb = out + BATCH + 3 * ray;
    orgb[0] = sigmoidf_(rr);
    orgb[1] = sigmoidf_(gg);
    orgb[2] = sigmoidf_(bb);
  }
}

extern "C" void kernel_launch(void* const* d_in, const int* in_sizes, int n_in,
                              void* d_out, int out_size, void* d_ws, size_t ws_size,
                              hipStream_t stream) {
  (void)in_sizes; (void)n_in; (void)out_size; (void)d_ws; (void)ws_size;
  const float* x  = (const float*)d_in[0];
  const float* g0 = (const float*)d_in[1];
  const float* g1 = (const float*)d_in[2];
  const float* g2 = (const float*)d_in[3];
  const float* wl = (const float*)d_in[4];
  const float* bl = (const float*)d_in[5];
  const float* wr = (const float*)d_in[6];
  const float* br = (const float*)d_in[7];
  float* out = (float*)d_out;

  // CDNA5 WGP has 320KB LDS; allow our 166KB dynamic allocation.
  (void)hipFuncSetAttribute((const void*)nerf_ray_march_kernel,
                            hipFuncAttributeMaxDynamicSharedMemorySize, LDS_BYTES);
  nerf_ray_march_kernel<<<NBLOCKS, BLOCK, LDS_BYTES, stream>>>(
      x, g0, g1, g2, wl, bl, wr, br, out);
}